// MonteCarloPooling_19653770346999
// MI455X (gfx1250) — compile-verified
//
#include <hip/hip_runtime.h>
#include <hip/hip_bf16.h>

// Monte-Carlo 2x2 pooling, MI455X (gfx1250).
// Pure streaming op: ~925 MB traffic, zero reuse -> bandwidth bound (~40us at
// 23.3 TB/s). Strategy: b128 non-temporal loads/stores (don't pollute 192MB L2),
// 4 outputs per lane for full coalescing, hardware v_log_f32 only (argmax is
// invariant under the log-base change and under removing the outer logs:
//   argmax ln(b)+gumbel(u) == argmax b / (-log2 u), all positive),
// cross-multiply tournament instead of divides. No matrix structure -> no WMMA.

typedef __attribute__((ext_vector_type(4))) float v4f;

#define POOL_EPS 1e-12f

__device__ __forceinline__ float gumbel_argmax4(float b0, float b1,
                                                float b2, float b3, v4f uu)
{
    const float lo = POOL_EPS;
    const float hi = 1.0f - 1e-12f;   // rounds to 1.0f in fp32, kept for fidelity

    // exponential draws (common positive scale 1/ln2 does not change the argmax)
    float e0 = -__builtin_amdgcn_logf(fminf(fmaxf(uu[0], lo), hi));
    float e1 = -__builtin_amdgcn_logf(fminf(fmaxf(uu[1], lo), hi));
    float e2 = -__builtin_amdgcn_logf(fminf(fmaxf(uu[2], lo), hi));
    float e3 = -__builtin_amdgcn_logf(fminf(fmaxf(uu[3], lo), hi));

    b0 = fmaxf(b0, POOL_EPS);
    b1 = fmaxf(b1, POOL_EPS);
    b2 = fmaxf(b2, POOL_EPS);
    b3 = fmaxf(b3, POOL_EPS);

    // score_k = b_k / e_k ; compare via cross-multiplication (all positive).
    // Strict '>' for the higher index reproduces argmax's first-occurrence ties.
    bool  c01  = (b1 * e0) > (b0 * e1);
    float i01  = c01 ? 1.0f : 0.0f;
    float bw01 = c01 ? b1 : b0;
    float ew01 = c01 ? e1 : e0;

    bool  c23  = (b3 * e2) > (b2 * e3);
    float i23  = c23 ? 3.0f : 2.0f;
    float bw23 = c23 ? b3 : b2;
    float ew23 = c23 ? e3 : e2;

    bool cf = (bw23 * ew01) > (bw01 * ew23);
    return cf ? i23 : i01;
}

__global__ __launch_bounds__(256) void
MonteCarloPooling_19653770346999_kernel(const float* __restrict__ x,
                                        const float* __restrict__ u,
                                        float* __restrict__ out, int nq)
{
    const int W   = 224;        // input width
    const int PW  = 112;        // pooled width
    const int PHW = 112 * 112;  // pooled plane (per B*C slice)

    int stride = (int)(gridDim.x * blockDim.x);
    for (int q = (int)(blockIdx.x * blockDim.x + threadIdx.x); q < nq; q += stride)
    {
        int oi = q << 2;                      // 4 pooled outputs, same pooled row
        unsigned bc  = (unsigned)oi / (unsigned)PHW;
        unsigned rem = (unsigned)oi - bc * (unsigned)PHW;
        unsigned ph  = rem / (unsigned)PW;
        unsigned pwb = rem - ph * (unsigned)PW;   // multiple of 4

        const float* xr0 = x + ((size_t)bc * 224 + 2 * (size_t)ph) * W + 2 * pwb;
        const float* xr1 = xr0 + W;
        const v4f*   up  = (const v4f*)(u + (size_t)oi * 4);

        // speculative prefetch of next grid-stride iteration into L2
        int qn = q + stride;
        if (qn < nq) {
            int oin = qn << 2;
            unsigned bcn  = (unsigned)oin / (unsigned)PHW;
            unsigned remn = (unsigned)oin - bcn * (unsigned)PHW;
            unsigned phn  = remn / (unsigned)PW;
            unsigned pwbn = remn - phn * (unsigned)PW;
            const float* xr0n = x + ((size_t)bcn * 224 + 2 * (size_t)phn) * W + 2 * pwbn;
            __builtin_prefetch(xr0n, 0, 1);
            __builtin_prefetch(xr0n + W, 0, 1);
            __builtin_prefetch(u + (size_t)oin * 4, 0, 1);
        }

        // two 32B strips of x (rows of the 2x2 blocks), non-temporal
        v4f r0a = __builtin_nontemporal_load((const v4f*)xr0);
        v4f r0b = __builtin_nontemporal_load((const v4f*)(xr0 + 4));
        v4f r1a = __builtin_nontemporal_load((const v4f*)xr1);
        v4f r1b = __builtin_nontemporal_load((const v4f*)(xr1 + 4));

        v4f u0 = __builtin_nontemporal_load(up + 0);
        v4f u1 = __builtin_nontemporal_load(up + 1);
        v4f u2 = __builtin_nontemporal_load(up + 2);
        v4f u3 = __builtin_nontemporal_load(up + 3);

        // block flatten order is (row, col): k = 2*row + col
        v4f res;
        res[0] = gumbel_argmax4(r0a[0], r0a[1], r1a[0], r1a[1], u0);
        res[1] = gumbel_argmax4(r0a[2], r0a[3], r1a[2], r1a[3], u1);
        res[2] = gumbel_argmax4(r0b[0], r0b[1], r1b[0], r1b[1], u2);
        res[3] = gumbel_argmax4(r0b[2], r0b[3], r1b[2], r1b[3], u3);

        __builtin_nontemporal_store(res, (v4f*)(out + oi));
    }
}

extern "C" void kernel_launch(void* const* d_in, const int* in_sizes, int n_in,
                              void* d_out, int out_size, void* d_ws, size_t ws_size,
                              hipStream_t stream)
{
    const float* x   = (const float*)d_in[0];   // (32,64,224,224) fp32
    const float* u   = (const float*)d_in[1];   // (32,64,112,112,4) fp32
    float*       out = (float*)d_out;           // (32,64,112,112) fp32 (indices)

    int nq = out_size >> 2;                     // 4 outputs per thread
    int block = 256;                            // 8 wave32 waves
    int grid  = (nq + block - 1) / block;
    if (grid > 4096) grid = 4096;               // grid-stride loop covers the rest

    MonteCarloPooling_19653770346999_kernel<<<grid, block, 0, stream>>>(x, u, out, nq);
}